// CortexNetwork_11785390260660
// MI455X (gfx1250) — compile-verified
//
#include <hip/hip_runtime.h>

typedef __attribute__((ext_vector_type(2))) float v2f;
typedef __attribute__((ext_vector_type(8))) float v8f;

#define NN   9216      // N = 96*96
#define BB   8         // batch
#define HH   119       // input H = 96-1+24
#define RFD  24
#define GXD  96
#define KCH  16        // K split factor
#define KLEN (NN / KCH)   // 576
#define NT   (NN / 16)    // 576 n-tiles of 16

// ---------------------------------------------------------------------------
// Lateral GEMM: D[b, n] = sum_k prev[b,k] * W[n,k]  for W_e and W_i at once.
// One wave per (n-tile of 16, k-chunk). fp32 WMMA 16x16x4:
//   A (16x4):  lane l holds prev[(l&15), k0 + 2*(l>>4) + {0,1}]
//   B (4x16):  lane l holds W[tile*16 + (l&15)][k0 + 2*(l>>4) + {0,1}]
// Both are a single per-lane b64 load in exactly the ISA register layout.
// A's pad rows (8..15) read clamped-but-garbage data; they only affect D rows
// 8..15 which are never stored, so no masking is needed.
// ---------------------------------------------------------------------------
__global__ __launch_bounds__(256) void lat_gemm_kernel(
    const float* __restrict__ prev,     // [8, 9216]
    const float* __restrict__ We,       // [9216, 9216] row n, col k
    const float* __restrict__ Wi,       // [9216, 9216]
    float* __restrict__ partial)        // [KCH][8][9216]
{
  const int wid  = (blockIdx.x * blockDim.x + threadIdx.x) >> 5;
  const int lane = threadIdx.x & 31;
  const int tile = wid % NT;
  const int kc   = wid / NT;

  const int  nsub  = lane & 15;
  const int  khalf = (lane >> 4) << 1;          // 0 or 2
  const long koff  = (long)kc * KLEN + khalf;

  const int brow = nsub < BB ? nsub : (BB - 1);  // clamp: safe address for pads

  const float* ap = prev + (long)brow * NN + koff;
  const float* ep = We  + (long)(tile * 16 + nsub) * NN + koff;
  const float* ip = Wi  + (long)(tile * 16 + nsub) * NN + koff;

  v8f ce = {};
  v8f ci = {};

  #pragma unroll 8
  for (int k = 0; k < KLEN; k += 4) {
    v2f a  = *(const v2f*)(ap + k);
    v2f be = __builtin_nontemporal_load((const v2f*)(ep + k));  // single-use stream
    v2f bi = __builtin_nontemporal_load((const v2f*)(ip + k));
    // 8 args: (neg_a, A, neg_b, B, c_mod, C, reuse_a, reuse_b)
    ce = __builtin_amdgcn_wmma_f32_16x16x4_f32(false, a, false, be, (short)0, ce,
                                               false, false);
    ci = __builtin_amdgcn_wmma_f32_16x16x4_f32(false, a, false, bi, (short)0, ci,
                                               false, false);
  }

  // D layout: VGPR r, lanes 0-15 -> (M=r, N=tile*16+lane). Rows 0..7 = batches.
  if (lane < 16) {
    const int n = tile * 16 + lane;
    float* p = partial + (long)kc * BB * NN + n;
    #pragma unroll
    for (int r = 0; r < BB; ++r)
      p[(long)r * NN] = 0.9f * (ce[r] - ci[r]);   // GAMMA_E*lat_e + GAMMA_I*lat_i
  }
}

// ---------------------------------------------------------------------------
// Finalize: afferent RF dot product + reduce K-chunk partials + ReLU.
// Thread layout idx = n*8 + b: the 8 lanes of one neuron share its 576 weights
// (broadcast loads), input patches overlap heavily (L2 resident).
// ---------------------------------------------------------------------------
__global__ __launch_bounds__(256) void finalize_kernel(
    const float* __restrict__ input,    // [8, 119, 119]
    const float* __restrict__ aw,       // [96, 96, 24, 24]
    const float* __restrict__ partial,  // [KCH][8][9216]
    float* __restrict__ out)            // [8, 9216]
{
  const int idx = blockIdx.x * blockDim.x + threadIdx.x;
  if (idx >= BB * NN) return;
  const int b = idx & 7;
  const int n = idx >> 3;
  const int i = n / GXD;
  const int j = n - i * GXD;

  const float* inp = input + (long)b * HH * HH;
  const float* w   = aw + (long)n * RFD * RFD;

  float acc = 0.0f;
  for (int h = 0; h < RFD; ++h) {
    const float* row = inp + (long)(i + h) * HH + j;
    const float* wr  = w + h * RFD;
    #pragma unroll
    for (int x = 0; x < RFD; ++x)
      acc = fmaf(row[x], wr[x], acc);
  }

  float lat = 0.0f;
  #pragma unroll
  for (int kc = 0; kc < KCH; ++kc)
    lat += partial[((long)kc * BB + b) * NN + n];

  out[(long)b * NN + n] = fmaxf(acc + lat, 0.0f);
}

extern "C" void kernel_launch(void* const* d_in, const int* in_sizes, int n_in,
                              void* d_out, int out_size, void* d_ws, size_t ws_size,
                              hipStream_t stream) {
  const float* input = (const float*)d_in[0];   // (B, H, H)
  const float* prev  = (const float*)d_in[1];   // (B, N)
  const float* aw    = (const float*)d_in[2];   // (GX, GY, RF, RF)
  const float* We    = (const float*)d_in[3];   // (N, N)
  const float* Wi    = (const float*)d_in[4];   // (N, N)
  // d_in[5], d_in[6] (rf_x, rf_y) are identity aranges -> folded into indexing.

  float* out     = (float*)d_out;
  float* partial = (float*)d_ws;                // KCH*8*NN floats = 4.7 MB

  const int gemm_waves  = NT * KCH;             // 9216 waves
  const int gemm_blocks = gemm_waves / 8;       // 256 threads = 8 waves per block
  lat_gemm_kernel<<<gemm_blocks, 256, 0, stream>>>(prev, We, Wi, partial);

  const int fin_threads = BB * NN;              // 73728
  finalize_kernel<<<(fin_threads + 255) / 256, 256, 0, stream>>>(input, aw,
                                                                 partial, out);
}